// HyperbolicGraphConvolution_62508954026440
// MI455X (gfx1250) — compile-verified
//
#include <hip/hip_runtime.h>
#include <math.h>

// ---------------------------------------------------------------------------
// Hyperbolic graph convolution (Poincare ball, c = 1) for MI455X / gfx1250.
//   K0  zero num/den accumulators
//   K1  per-node: gamma, gamma*x, gamma-1
//   K2  edge scatter: float4 gather + fp32 global atomics (L2-resident)
//   K3  gyro-midpoint finalize + residual midpoint with h_init (row per wave)
//   K4  WMMA fp32 GEMM: mx = support @ W^T  (v_wmma_f32_16x16x4_f32, W in LDS)
//   K5  mobius_matvec rescale + mobius_add(expmap0(bias)) + project
// ---------------------------------------------------------------------------

#define DDIM 48
#define LSTRIDE 50  // LDS row stride (floats): even (b64-aligned), bank-conflict free
#define EPSF 1e-15f
#define MAXNORM (1.0f - 1e-5f)
#define ATANH_CLIP 0.9999999f
#define CLAMP_ABS_EPS 1e-10f
#define ALPHA 0.1f

typedef __attribute__((ext_vector_type(2))) float v2f;
typedef __attribute__((ext_vector_type(8))) float v8f;

__device__ __forceinline__ float wave_sum(float v) {
#pragma unroll
  for (int off = 16; off > 0; off >>= 1) v += __shfl_xor(v, off, 32);
  return v;
}

__device__ __forceinline__ float artanh_clipped(float x) {
  float t = fminf(fmaxf(x, -ATANH_CLIP), ATANH_CLIP);
  return 0.5f * logf((1.0f + t) / (1.0f - t));
}

__device__ __forceinline__ float clamp_abs(float x) {
  return (x >= 0.0f) ? fmaxf(x, CLAMP_ABS_EPS) : fminf(x, -CLAMP_ABS_EPS);
}

// mobius_scalar_mul(0.5, u) followed by _project, on the (a,b) lane pair.
// Reductions are wave-wide; lanes >= 16 carry b = 0 padding (dim 48..63).
__device__ __forceinline__ void half_scalar_mul_project(float& a, float& b) {
  float n2 = wave_sum(a * a + b * b);
  float n = fmaxf(sqrtf(n2), EPSF);
  float sc = tanhf(0.5f * artanh_clipped(n)) / n;
  a *= sc;
  b *= sc;
  float r2 = wave_sum(a * a + b * b);
  float rn = sqrtf(r2);
  if (rn > MAXNORM) {
    float f = MAXNORM / fmaxf(rn, EPSF);
    a *= f;
    b *= f;
  }
}

// --------------------------- K0: zero accumulators --------------------------
__global__ void __launch_bounds__(256) k0_zero(float* p, long long n) {
  long long i = (long long)blockIdx.x * blockDim.x + threadIdx.x;
  if (i < n) p[i] = 0.0f;
}

// --------------------------- K1: per-node precompute ------------------------
__global__ void __launch_bounds__(256) k1_precompute(const float* __restrict__ x,
                                                     float* __restrict__ gx,
                                                     float* __restrict__ gm1, int N) {
  int row = (blockIdx.x * blockDim.x + threadIdx.x) >> 5;
  int lane = threadIdx.x & 31;
  if (row >= N) return;
  const float* xr = x + (size_t)row * DDIM;
  float a = xr[lane];
  float b = (lane < 16) ? xr[lane + 32] : 0.0f;
  float s = wave_sum(a * a + b * b);
  float gamma = 2.0f / fmaxf(1.0f - s, EPSF);
  float* gr = gx + (size_t)row * DDIM;
  gr[lane] = gamma * a;
  if (lane < 16) gr[lane + 32] = gamma * b;
  if (lane == 0) gm1[row] = gamma - 1.0f;
}

// --------------------------- K2: edge scatter (atomics) ---------------------
// 12 threads per edge; each handles 4 contiguous dims via a b128 gather.
__global__ void __launch_bounds__(256) k2_edges(const float* __restrict__ gx,
                                                const float* __restrict__ gm1,
                                                const float* __restrict__ w,
                                                const int* __restrict__ src,
                                                const int* __restrict__ dst,
                                                float* __restrict__ num,
                                                float* __restrict__ den, long long total) {
  long long tid = (long long)blockIdx.x * blockDim.x + threadIdx.x;
  if (tid >= total) return;
  int e = (int)(tid / 12);
  int d4 = (int)(tid - (long long)e * 12) * 4;
  int s = src[e];
  int t = dst[e];
  float we = w[e];
  const float4 g = *(const float4*)(gx + (size_t)s * DDIM + d4);
  float* np = num + (size_t)t * DDIM + d4;
  unsafeAtomicAdd(np + 0, we * g.x);
  unsafeAtomicAdd(np + 1, we * g.y);
  unsafeAtomicAdd(np + 2, we * g.z);
  unsafeAtomicAdd(np + 3, we * g.w);
  if (d4 == 0) unsafeAtomicAdd(&den[t], fabsf(we) * gm1[s]);
}

// ------------------ K3: midpoint finalize + residual midpoint ---------------
__global__ void __launch_bounds__(256) k3_midpoint(const float* __restrict__ num,
                                                   const float* __restrict__ den,
                                                   const float* __restrict__ h_init,
                                                   float* __restrict__ support, int N) {
  int row = (blockIdx.x * blockDim.x + threadIdx.x) >> 5;
  int lane = threadIdx.x & 31;
  if (row >= N) return;

  float dv = clamp_abs(den[row]);
  const float* nr = num + (size_t)row * DDIM;
  float ua = nr[lane] / dv;
  float ub = (lane < 16) ? nr[lane + 32] / dv : 0.0f;
  half_scalar_mul_project(ua, ub);  // -> support_spmm (point a)

  const float* hr = h_init + (size_t)row * DDIM;
  float ha = hr[lane];
  float hb = (lane < 16) ? hr[lane + 32] : 0.0f;

  float a2 = wave_sum(ua * ua + ub * ub);
  float b2 = wave_sum(ha * ha + hb * hb);
  float ga = 2.0f / fmaxf(1.0f - a2, EPSF);
  float gb = 2.0f / fmaxf(1.0f - b2, EPSF);
  const float wa = 1.0f - ALPHA, wb = ALPHA;
  float d2 = clamp_abs(wa * (ga - 1.0f) + wb * (gb - 1.0f));
  float va = (wa * ga * ua + wb * gb * ha) / d2;
  float vb = (wa * ga * ub + wb * gb * hb) / d2;
  half_scalar_mul_project(va, vb);

  float* sr = support + (size_t)row * DDIM;
  sr[lane] = va;
  if (lane < 16) sr[lane + 32] = vb;
}

// --------------------------- K4: WMMA GEMM  mx = S @ W^T --------------------
// One wave per 16-row tile; 3 j-tiles (48 cols); K in steps of 4.
// W staged in LDS (padded stride 50 -> no bank conflicts, b64-aligned).
__global__ void __launch_bounds__(256) k4_wmma_gemm(const float* __restrict__ S,
                                                    const float* __restrict__ W,
                                                    float* __restrict__ mx, int N) {
  __shared__ float Wl[DDIM * LSTRIDE];
  for (int i = threadIdx.x; i < DDIM * DDIM; i += 256) {
    int r = i / DDIM;
    int cidx = i - r * DDIM;
    Wl[r * LSTRIDE + cidx] = W[i];  // Wl[j][k] = W[j][k]
  }
  __syncthreads();

  int lane = threadIdx.x & 31;
  int tile = blockIdx.x * (blockDim.x >> 5) + (threadIdx.x >> 5);
  int m0 = tile * 16;
  if (m0 >= N) return;  // wave-uniform: EXEC stays all-ones for WMMA

  int kh = lane >> 4;  // 0: K+{0,1}   1: K+{2,3}
  int l15 = lane & 15;
  int rowA = m0 + l15;

  const float* b0p = Wl + (size_t)(0 + l15) * LSTRIDE + kh * 2;
  const float* b1p = Wl + (size_t)(16 + l15) * LSTRIDE + kh * 2;
  const float* b2p = Wl + (size_t)(32 + l15) * LSTRIDE + kh * 2;

  v8f c0 = {}, c1 = {}, c2 = {};

  if (m0 + 16 <= N) {
    // Full tile: no per-lane guards -> clean load clauses + WMMA.
    const float* ap = S + (size_t)rowA * DDIM + kh * 2;
#pragma unroll
    for (int k = 0; k < DDIM; k += 4) {
      v2f a, b0, b1, b2;
      a.x = ap[k];
      a.y = ap[k + 1];
      b0.x = b0p[k]; b0.y = b0p[k + 1];
      b1.x = b1p[k]; b1.y = b1p[k + 1];
      b2.x = b2p[k]; b2.y = b2p[k + 1];
      c0 = __builtin_amdgcn_wmma_f32_16x16x4_f32(false, a, false, b0, (short)0, c0, false, false);
      c1 = __builtin_amdgcn_wmma_f32_16x16x4_f32(false, a, false, b1, (short)0, c1, false, false);
      c2 = __builtin_amdgcn_wmma_f32_16x16x4_f32(false, a, false, b2, (short)0, c2, false, false);
    }
#pragma unroll
    for (int v = 0; v < 8; ++v) {
      float* o = mx + (size_t)(m0 + v + kh * 8) * DDIM + l15;
      o[0] = c0[v];
      o[16] = c1[v];
      o[32] = c2[v];
    }
  } else {
    // Tail tile: guard OOB rows (compiler restores EXEC before each WMMA).
    bool rowOK = rowA < N;
#pragma unroll
    for (int k = 0; k < DDIM; k += 4) {
      v2f a, b0, b1, b2;
      if (rowOK) {
        const float* p = S + (size_t)rowA * DDIM + kh * 2 + k;
        a.x = p[0];
        a.y = p[1];
      } else {
        a.x = 0.0f;
        a.y = 0.0f;
      }
      b0.x = b0p[k]; b0.y = b0p[k + 1];
      b1.x = b1p[k]; b1.y = b1p[k + 1];
      b2.x = b2p[k]; b2.y = b2p[k + 1];
      c0 = __builtin_amdgcn_wmma_f32_16x16x4_f32(false, a, false, b0, (short)0, c0, false, false);
      c1 = __builtin_amdgcn_wmma_f32_16x16x4_f32(false, a, false, b1, (short)0, c1, false, false);
      c2 = __builtin_amdgcn_wmma_f32_16x16x4_f32(false, a, false, b2, (short)0, c2, false, false);
    }
#pragma unroll
    for (int v = 0; v < 8; ++v) {
      int row = m0 + v + kh * 8;
      if (row < N) {
        float* o = mx + (size_t)row * DDIM + l15;
        o[0] = c0[v];
        o[16] = c1[v];
        o[32] = c2[v];
      }
    }
  }
}

// ---------- K5: mobius_matvec rescale + mobius_add(expmap0(bias)) -----------
__global__ void __launch_bounds__(256) k5_epilogue(const float* __restrict__ S,
                                                   const float* __restrict__ mx,
                                                   const float* __restrict__ bias,
                                                   float* __restrict__ out, int N) {
  int row = (blockIdx.x * blockDim.x + threadIdx.x) >> 5;
  int lane = threadIdx.x & 31;
  if (row >= N) return;

  const float* sr = S + (size_t)row * DDIM;
  float sa = sr[lane];
  float sb = (lane < 16) ? sr[lane + 32] : 0.0f;
  float xn = fmaxf(sqrtf(wave_sum(sa * sa + sb * sb)), EPSF);

  const float* mr = mx + (size_t)row * DDIM;
  float ma = mr[lane];
  float mb = (lane < 16) ? mr[lane + 32] : 0.0f;
  float mxn = fmaxf(sqrtf(wave_sum(ma * ma + mb * mb)), EPSF);

  // res = tanh(mxn/xn * artanh(xn)) * mx / mxn ; then project
  float sc = tanhf((mxn / xn) * artanh_clipped(xn)) / mxn;
  float ra = sc * ma, rb = sc * mb;
  float rn = sqrtf(wave_sum(ra * ra + rb * rb));
  if (rn > MAXNORM) {
    float f = MAXNORM / fmaxf(rn, EPSF);
    ra *= f;
    rb *= f;
  }

  // b0 = expmap0(bias)
  float ba = bias[lane];
  float bb = (lane < 16) ? bias[lane + 32] : 0.0f;
  float bn = fmaxf(sqrtf(wave_sum(ba * ba + bb * bb)), EPSF);
  float bsc = tanhf(bn) / bn;
  float b0a = bsc * ba, b0b = bsc * bb;

  // mobius_add(res, b0)
  float x2 = wave_sum(ra * ra + rb * rb);
  float y2 = wave_sum(b0a * b0a + b0b * b0b);
  float xy = wave_sum(ra * b0a + rb * b0b);
  float ca = 1.0f + 2.0f * xy + y2;
  float cb = 1.0f - x2;
  float dn = fmaxf(1.0f + 2.0f * xy + x2 * y2, EPSF);
  float ha = (ca * ra + cb * b0a) / dn;
  float hb = (ca * rb + cb * b0b) / dn;

  float hn = sqrtf(wave_sum(ha * ha + hb * hb));
  if (hn > MAXNORM) {
    float f = MAXNORM / fmaxf(hn, EPSF);
    ha *= f;
    hb *= f;
  }

  float* o = out + (size_t)row * DDIM;
  o[lane] = ha;
  if (lane < 16) o[lane + 32] = hb;
}

// ---------------------------------------------------------------------------
extern "C" void kernel_launch(void* const* d_in, const int* in_sizes, int n_in,
                              void* d_out, int out_size, void* d_ws, size_t ws_size,
                              hipStream_t stream) {
  const float* x = (const float*)d_in[0];
  const float* h_init = (const float*)d_in[1];
  const float* edge_w = (const float*)d_in[2];
  const float* W = (const float*)d_in[3];
  const float* bias = (const float*)d_in[4];
  const int* edge_src = (const int*)d_in[5];
  const int* edge_dst = (const int*)d_in[6];
  float* out = (float*)d_out;

  const int N = in_sizes[0] / DDIM;
  const long long E = in_sizes[2];

  // Workspace layout (floats): gx[N*48] | gm1[N] | num[N*48] | den[N] |
  //                            support[N*48] | mx[N*48]
  float* ws = (float*)d_ws;
  float* gx = ws;
  float* gm1 = gx + (size_t)N * DDIM;
  float* num = gm1 + N;  // num & den contiguous -> single zero pass
  float* den = num + (size_t)N * DDIM;
  float* support = den + N;
  float* mxb = support + (size_t)N * DDIM;

  const int TB = 256;

  // K0: zero num+den (N*49 contiguous floats)
  long long nz = (long long)N * (DDIM + 1);
  k0_zero<<<(unsigned)((nz + TB - 1) / TB), TB, 0, stream>>>(num, nz);

  // K1: per-node precompute (one wave per row)
  int waveBlocks = (N + (TB / 32) - 1) / (TB / 32);
  k1_precompute<<<waveBlocks, TB, 0, stream>>>(x, gx, gm1, N);

  // K2: edge scatter, 12 threads per edge (4 dims each, b128 gather)
  long long total = E * 12;
  k2_edges<<<(unsigned)((total + TB - 1) / TB), TB, 0, stream>>>(
      gx, gm1, edge_w, edge_src, edge_dst, num, den, total);

  // K3: midpoint finalize + residual midpoint
  k3_midpoint<<<waveBlocks, TB, 0, stream>>>(num, den, h_init, support, N);

  // K4: WMMA GEMM (one wave per 16-row tile, 8 waves per block)
  int tiles = (N + 15) / 16;
  int gemmBlocks = (tiles + (TB / 32) - 1) / (TB / 32);
  k4_wmma_gemm<<<gemmBlocks, TB, 0, stream>>>(support, W, mxb, N);

  // K5: epilogue
  k5_epilogue<<<waveBlocks, TB, 0, stream>>>(support, mxb, bias, out, N);
}